// Neighborhood3DAttentionLayer_ResStyle_37460704756338
// MI455X (gfx1250) — compile-verified
//
#include <hip/hip_runtime.h>
#include <math.h>

// ---------------- problem constants ----------------
constexpr int Tn = 3, Hn = 32, Wn = 32, Cn = 384, NHn = 4, DHn = 96, EMBn = 512;
constexpr int Gn = 32, CPG = Cn / Gn;          // 12 channels / group
constexpr int HW = Hn * Wn;                    // 1024
constexpr int SP = Tn * HW;                    // 3072 spatial positions
constexpr int C3 = 3 * Cn;                     // 1152
constexpr int KTOT = Cn * 27;                  // 10368 implicit-GEMM K for conv
constexpr float EPSc = 1e-5f;
constexpr float SCALEc = 0.10206207261596577f; // 96^-0.5

// ---------------- WMMA f32 16x16x4 ----------------
typedef __attribute__((ext_vector_type(2))) float v2f;
typedef __attribute__((ext_vector_type(8))) float v8f;

static __device__ __forceinline__ v8f wmma4(v2f a, v2f b, v8f c) {
  return __builtin_amdgcn_wmma_f32_16x16x4_f32(false, a, false, b, (short)0, c,
                                               false, false);
}

// ---------------- wave32 helpers ----------------
static __device__ __forceinline__ float wave_sum(float v) {
  for (int off = 16; off; off >>= 1) v += __shfl_xor(v, off, 32);
  return v;
}
static __device__ __forceinline__ float wave_max(float v) {
  for (int off = 16; off; off >>= 1) v = fmaxf(v, __shfl_xor(v, off, 32));
  return v;
}
static __device__ __forceinline__ float silu(float x) {
  return x / (1.0f + expf(-x));
}

// ============ 1) GroupNorm-in + SiLU -> xf[s][c] ============
__global__ void k_gn_in(const float* __restrict__ x, const float* __restrict__ w,
                        const float* __restrict__ b, float* __restrict__ xf) {
  const int g = blockIdx.x, tid = threadIdx.x;
  __shared__ float ssum[256], ssq[256];
  float s = 0.f, q = 0.f;
  for (int i = tid; i < CPG * SP; i += 256) {
    int cc = i / SP, sp = i % SP;
    int c = g * CPG + cc;
    int t = sp >> 10, hw = sp & 1023;
    float v = x[(t * Cn + c) * HW + hw];
    s += v; q += v * v;
  }
  ssum[tid] = s; ssq[tid] = q; __syncthreads();
  for (int st = 128; st; st >>= 1) {
    if (tid < st) { ssum[tid] += ssum[tid + st]; ssq[tid] += ssq[tid + st]; }
    __syncthreads();
  }
  const float inv_n = 1.0f / (float)(CPG * SP);
  float mu = ssum[0] * inv_n;
  float var = ssq[0] * inv_n - mu * mu;
  float rs = rsqrtf(var + EPSc);
  for (int i = tid; i < CPG * SP; i += 256) {
    int cc = i / SP, sp = i % SP;
    int c = g * CPG + cc;
    int t = sp >> 10, hw = sp & 1023;
    float v = x[(t * Cn + c) * HW + hw];
    float xn = (v - mu) * rs * w[c] + b[c];
    xf[sp * Cn + c] = silu(xn);
  }
}

// ============ 2) QKV GEMM: 16x64 strip per wave, D[s][n] = xf @ w_qkv^T ============
__global__ void k_gemm_qkv(const float* __restrict__ A, const float* __restrict__ Wq,
                           float* __restrict__ D) {
  const int bid = blockIdx.x;
  const int m0 = (bid % (SP / 16)) * 16;
  const int n0 = (bid / (SP / 16)) * 64;     // 4 N-tiles per wave
  const int lane = threadIdx.x & 31;
  const int half = lane >> 4, mn = lane & 15;
  v8f acc[4] = {{}, {}, {}, {}};
  const float* arow = A + (m0 + mn) * Cn;
  const float* bp[4];
#pragma unroll
  for (int u = 0; u < 4; ++u) bp[u] = Wq + (n0 + 16 * u + mn) * Cn;
  __builtin_prefetch(arow, 0, 3);
  __builtin_prefetch(bp[0], 0, 3);
  for (int k0 = 0; k0 < Cn; k0 += 4) {
    int kb = k0 + half * 2;
    v2f a;
    a.x = arow[kb]; a.y = arow[kb + 1];
#pragma unroll
    for (int u = 0; u < 4; ++u) {
      v2f b;
      b.x = bp[u][kb]; b.y = bp[u][kb + 1];
      acc[u] = wmma4(a, b, acc[u]);
    }
  }
#pragma unroll
  for (int u = 0; u < 4; ++u)
#pragma unroll
    for (int r = 0; r < 8; ++r) {
      int row = m0 + r + half * 8;
      D[row * C3 + n0 + 16 * u + mn] = acc[u][r];
    }
}

// ============ 3) per-head LayerNorm + RoPE3D (+scale) & V relayout ============
__global__ void k_ln_rope(const float* __restrict__ qkv,
                          const float* __restrict__ qnw, const float* __restrict__ knw,
                          float* __restrict__ qo, float* __restrict__ ko,
                          float* __restrict__ vo) {
  const int wave = (blockIdx.x * blockDim.x + threadIdx.x) >> 5;
  const int lane = threadIdx.x & 31;
  if (wave >= NHn * SP) return;
  const int nh = wave / SP, s = wave % SP;
  const int t = s >> 10, h = (s >> 5) & 31, w = s & 31;
  const float invf = expf(-(float)(lane & 15) * 0.5756462732485114f); // 10000^(-i/16)
  const float pos[3] = {(float)t, (float)h, (float)w};

  for (int which = 0; which < 2; ++which) { // 0=q, 1=k
    const float* row = qkv + s * C3 + which * Cn + nh * DHn;
    const float* lw = which ? knw : qnw;
    float v0 = row[lane], v1 = row[lane + 32], v2 = row[lane + 64];
    float mu = wave_sum(v0 + v1 + v2) * (1.0f / 96.0f);
    float var = wave_sum(v0 * v0 + v1 * v1 + v2 * v2) * (1.0f / 96.0f) - mu * mu;
    float rs = rsqrtf(var + EPSc);
    float xn[3] = {(v0 - mu) * rs * lw[lane],
                   (v1 - mu) * rs * lw[lane + 32],
                   (v2 - mu) * rs * lw[lane + 64]};
    float* dst = (which ? ko : qo) + (nh * SP + s) * DHn;
    float mul = which ? 1.0f : SCALEc;
    for (int e = 0; e < 3; ++e) {
      float ang = pos[e] * invf;
      float cv = cosf(ang), sv = sinf(ang);
      float other = __shfl_xor(xn[e], 16, 32);
      float o = (lane < 16) ? (xn[e] * cv - other * sv)
                            : (other * sv + xn[e] * cv);
      dst[e * 32 + lane] = o * mul;
    }
    if (which == 1) { // also copy v on second pass
      const float* vr = qkv + s * C3 + 2 * Cn + nh * DHn;
      float* vd = vo + (nh * SP + s) * DHn;
      vd[lane] = vr[lane]; vd[lane + 32] = vr[lane + 32]; vd[lane + 64] = vr[lane + 64];
    }
  }
}

// ============ 4) neighborhood attention: wave per (head, query) ============
__global__ void k_attn(const float* __restrict__ qo, const float* __restrict__ ko,
                       const float* __restrict__ vo, float* __restrict__ ao) {
  const int wave = (blockIdx.x * blockDim.x + threadIdx.x) >> 5;
  const int lane = threadIdx.x & 31;
  if (wave >= NHn * SP) return;
  const int nh = wave / SP, s = wave % SP;
  const int h = (s >> 5) & 31, w = s & 31;
  const int sh = min(max(h - 2, 0), Hn - 5);
  const int sw = min(max(w - 2, 0), Wn - 5);
  const float* qb = qo + (nh * SP + s) * DHn;
  const float q0 = qb[lane], q1 = qb[lane + 32], q2 = qb[lane + 64];
  float l0 = -1e30f, l1 = -1e30f, l2 = -1e30f;
  for (int j = 0; j < 75; ++j) {
    int jt = j / 25, r = j % 25, jh = r / 5, jw = r % 5;
    int sn = jt * HW + (sh + jh) * Wn + (sw + jw);
    const float* kb = ko + (nh * SP + sn) * DHn;
    float d = q0 * kb[lane] + q1 * kb[lane + 32] + q2 * kb[lane + 64];
    d = wave_sum(d);
    int reg = j >> 5, ln = j & 31;
    if (lane == ln) { if (reg == 0) l0 = d; else if (reg == 1) l1 = d; else l2 = d; }
  }
  float m = wave_max(fmaxf(l0, fmaxf(l1, l2)));
  float p0 = expf(l0 - m), p1 = expf(l1 - m), p2 = expf(l2 - m);
  float inv = 1.0f / wave_sum(p0 + p1 + p2);
  float a0 = 0.f, a1 = 0.f, a2 = 0.f;
  for (int j = 0; j < 75; ++j) {
    int jt = j / 25, r = j % 25, jh = r / 5, jw = r % 5;
    int sn = jt * HW + (sh + jh) * Wn + (sw + jw);
    int reg = j >> 5, ln = j & 31;
    float w0s = __shfl(p0, ln, 32), w1s = __shfl(p1, ln, 32), w2s = __shfl(p2, ln, 32);
    float wj = ((reg == 0) ? w0s : (reg == 1) ? w1s : w2s) * inv;
    const float* vb = vo + (nh * SP + sn) * DHn;
    a0 += wj * vb[lane]; a1 += wj * vb[lane + 32]; a2 += wj * vb[lane + 64];
  }
  float* dst = ao + s * Cn + nh * DHn;
  dst[lane] = a0; dst[lane + 32] = a1; dst[lane + 64] = a2;
}

// ============ 5) out-proj GEMM (16x64 strip) + skip: xr[s][c] ============
__global__ void k_gemm_out(const float* __restrict__ A, const float* __restrict__ Wo,
                           const float* __restrict__ x, float* __restrict__ xr) {
  const int bid = blockIdx.x;
  const int m0 = (bid % (SP / 16)) * 16;
  const int n0 = (bid / (SP / 16)) * 64;
  const int lane = threadIdx.x & 31;
  const int half = lane >> 4, mn = lane & 15;
  v8f acc[4] = {{}, {}, {}, {}};
  const float* arow = A + (m0 + mn) * Cn;
  const float* bp[4];
#pragma unroll
  for (int u = 0; u < 4; ++u) bp[u] = Wo + (n0 + 16 * u + mn) * Cn;
  __builtin_prefetch(arow, 0, 3);
  __builtin_prefetch(bp[0], 0, 3);
  for (int k0 = 0; k0 < Cn; k0 += 4) {
    int kb = k0 + half * 2;
    v2f a;
    a.x = arow[kb]; a.y = arow[kb + 1];
#pragma unroll
    for (int u = 0; u < 4; ++u) {
      v2f b;
      b.x = bp[u][kb]; b.y = bp[u][kb + 1];
      acc[u] = wmma4(a, b, acc[u]);
    }
  }
#pragma unroll
  for (int u = 0; u < 4; ++u)
#pragma unroll
    for (int r = 0; r < 8; ++r) {
      int row = m0 + r + half * 8;
      int c = n0 + 16 * u + mn;
      int t = row >> 10, hw = row & 1023;
      float skip = x[(t * Cn + c) * HW + hw];
      xr[row * Cn + c] = acc[u][r] + skip;
    }
}

// ============ 6) emb GEMM -> scale/shift per (c,t) ============
__global__ void k_emb(const float* __restrict__ emb, const float* __restrict__ wemb,
                      const float* __restrict__ bemb, float* __restrict__ mod) {
  int j = blockIdx.x * blockDim.x + threadIdx.x;
  if (j >= Tn * 2 * Cn) return;
  int tt = j / (2 * Cn), col = j % (2 * Cn);
  float s = bemb[col];
  const float* er = emb + tt * EMBn;
  const float* wr = wemb + col * EMBn;
  for (int e = 0; e < EMBn; ++e) s += er[e] * wr[e];
  if (col < Cn) mod[col * Tn + tt] = s;                 // scale
  else          mod[Cn * Tn + (col - Cn) * Tn + tt] = s; // shift
}

// ============ 7) GroupNorm-out + modulation + SiLU -> act[c][s] ============
__global__ void k_gn_out(const float* __restrict__ xr, const float* __restrict__ w,
                         const float* __restrict__ b, const float* __restrict__ mod,
                         float* __restrict__ act) {
  const int g = blockIdx.x, tid = threadIdx.x;
  __shared__ float ssum[256], ssq[256];
  float s = 0.f, q = 0.f;
  for (int i = tid; i < CPG * SP; i += 256) {
    int cc = i / SP, sp = i % SP;
    int c = g * CPG + cc;
    float v = xr[sp * Cn + c];
    s += v; q += v * v;
  }
  ssum[tid] = s; ssq[tid] = q; __syncthreads();
  for (int st = 128; st; st >>= 1) {
    if (tid < st) { ssum[tid] += ssum[tid + st]; ssq[tid] += ssq[tid + st]; }
    __syncthreads();
  }
  const float inv_n = 1.0f / (float)(CPG * SP);
  float mu = ssum[0] * inv_n;
  float var = ssq[0] * inv_n - mu * mu;
  float rs = rsqrtf(var + EPSc);
  for (int i = tid; i < CPG * SP; i += 256) {
    int cc = i / SP, sp = i % SP;
    int c = g * CPG + cc;
    int t = sp >> 10;
    float v = xr[sp * Cn + c];
    float y = (v - mu) * rs * w[c] + b[c];
    y = y * (1.0f + mod[c * Tn + t]) + mod[Cn * Tn + c * Tn + t];
    act[c * SP + sp] = silu(y);
  }
}

// ============ 8) Conv3d 3x3x3 SAME, implicit GEMM, 16 pos x 64 co per wave ============
// K-order restructured: outer 27 taps (kt,kh,kw), inner ci in chunks of 4.
// Logical k for the WMMA chunk = {ci..ci+3} at a fixed tap; A/B agree on it, and
// the sum over K is order independent, so this matches the flat conv reduction.
__global__ void k_conv(const float* __restrict__ act, const float* __restrict__ wc,
                       const float* __restrict__ bc, const float* __restrict__ x,
                       float* __restrict__ out) {
  const int bid = blockIdx.x;
  const int ng = bid % (Cn / 64);            // 6 co-groups of 64
  const int spi = bid / (Cn / 64);           // [0, 192): t(3) * h(32) * wtile(2)
  const int wt = spi & 1, h = (spi >> 1) & 31, t = spi >> 6;
  const int w0 = wt * 16;
  const int n0 = ng * 64;
  const int lane = threadIdx.x & 31;
  const int half = lane >> 4, mn = lane & 15;
  v8f acc[4] = {{}, {}, {}, {}};
  const float* bp[4];
#pragma unroll
  for (int u = 0; u < 4; ++u) bp[u] = wc + (n0 + 16 * u + mn) * KTOT;
  __builtin_prefetch(bp[0], 0, 3);

  for (int tap = 0; tap < 27; ++tap) {
    const int kt = tap / 9, r9 = tap - kt * 9;
    const int kh = r9 / 3, kw = r9 - kh * 3;
    const int t2 = t + kt - 1, h2 = h + kh - 1;
    const int w2 = w0 + mn + kw - 1;          // per-lane A row position
    const bool ok = ((unsigned)t2 < (unsigned)Tn) &&
                    ((unsigned)h2 < (unsigned)Hn) &&
                    ((unsigned)w2 < (unsigned)Wn);
    const int sbase = ok ? (t2 * HW + h2 * Wn + w2) : 0;
    const float* ab = act + sbase;            // + ci*SP per channel
    const float* wb0 = bp[0] + tap;           // + ci*27 per channel
    const float* wb1 = bp[1] + tap;
    const float* wb2 = bp[2] + tap;
    const float* wb3 = bp[3] + tap;
    for (int ci = 0; ci < Cn; ci += 4) {
      int cb = ci + half * 2;
      v2f a;
      a.x = ok ? ab[cb * SP] : 0.0f;
      a.y = ok ? ab[(cb + 1) * SP] : 0.0f;
      v2f b;
      b.x = wb0[cb * 27]; b.y = wb0[(cb + 1) * 27];
      acc[0] = wmma4(a, b, acc[0]);
      b.x = wb1[cb * 27]; b.y = wb1[(cb + 1) * 27];
      acc[1] = wmma4(a, b, acc[1]);
      b.x = wb2[cb * 27]; b.y = wb2[(cb + 1) * 27];
      acc[2] = wmma4(a, b, acc[2]);
      b.x = wb3[cb * 27]; b.y = wb3[(cb + 1) * 27];
      acc[3] = wmma4(a, b, acc[3]);
    }
  }
#pragma unroll
  for (int u = 0; u < 4; ++u)
#pragma unroll
    for (int r = 0; r < 8; ++r) {
      int m = r + half * 8;
      int co = n0 + 16 * u + mn;
      int idx = (t * Cn + co) * HW + h * Wn + (w0 + m);
      out[idx] = acc[u][r] + bc[co] + x[idx];
    }
}

// ---------------- host launcher ----------------
extern "C" void kernel_launch(void* const* d_in, const int* in_sizes, int n_in,
                              void* d_out, int out_size, void* d_ws, size_t ws_size,
                              hipStream_t stream) {
  const float* x        = (const float*)d_in[0];
  const float* emb      = (const float*)d_in[1];
  const float* w_qkv    = (const float*)d_in[2];
  const float* w_ao     = (const float*)d_in[3];
  const float* qn_w     = (const float*)d_in[4];
  const float* kn_w     = (const float*)d_in[5];
  const float* gn_in_w  = (const float*)d_in[6];
  const float* gn_in_b  = (const float*)d_in[7];
  const float* w_emb    = (const float*)d_in[8];
  const float* b_emb    = (const float*)d_in[9];
  const float* gn_out_w = (const float*)d_in[10];
  const float* gn_out_b = (const float*)d_in[11];
  const float* w_conv   = (const float*)d_in[12];
  const float* b_conv   = (const float*)d_in[13];
  float* out = (float*)d_out;

  float* ws = (float*)d_ws;
  float* xf   = ws;                        // SP*Cn     (reused as attn-out)
  float* qkv  = xf + SP * Cn;              // SP*C3     (reused as conv act)
  float* qb   = qkv + SP * C3;             // NH*SP*DH  (reused as xr)
  float* kb   = qb + NHn * SP * DHn;
  float* vb   = kb + NHn * SP * DHn;
  float* mod  = vb + NHn * SP * DHn;       // 2*Cn*Tn
  float* attn = xf;
  float* xr   = qb;
  float* act  = qkv;

  k_gn_in<<<Gn, 256, 0, stream>>>(x, gn_in_w, gn_in_b, xf);
  k_gemm_qkv<<<(SP / 16) * (C3 / 64), 32, 0, stream>>>(xf, w_qkv, qkv);
  k_ln_rope<<<(NHn * SP * 32) / 128, 128, 0, stream>>>(qkv, qn_w, kn_w, qb, kb, vb);
  k_attn<<<(NHn * SP * 32) / 128, 128, 0, stream>>>(qb, kb, vb, attn);
  k_gemm_out<<<(SP / 16) * (Cn / 64), 32, 0, stream>>>(attn, w_ao, x, xr);
  k_emb<<<(Tn * 2 * Cn + 255) / 256, 256, 0, stream>>>(emb, w_emb, b_emb, mod);
  k_gn_out<<<Gn, 256, 0, stream>>>(xr, gn_out_w, gn_out_b, mod, act);
  k_conv<<<(Tn * Hn * 2) * (Cn / 64), 32, 0, stream>>>(act, w_conv, b_conv, x, out);
}